// TreeDecoder_45303315038208
// MI455X (gfx1250) — compile-verified
//
#include <hip/hip_runtime.h>
#include <math.h>

// ---------------------------------------------------------------------------
// TreeDecoder forward, level-vectorized, for gfx1250 (MI455X).
// All GEMMs run through V_WMMA_F32_16X16X4_F32 (f32-exact matrix core path).
// Sizes (fixed by the reference): H=1024, E=1024, V+1=513, S=512, K=2, D=10.
// Register-blocked: RT=4 -> 64x64 C tile per wave = 16 WMMA per 8 B64 loads.
// ---------------------------------------------------------------------------

typedef __attribute__((ext_vector_type(2))) float v2f;
typedef __attribute__((ext_vector_type(8))) float v8f;

#define HD   1024      // hidden/embedding size
#define SD   512       // number of annotations
#define VP1  513       // V+1 logits
#define GD   4096      // 4*H LSTM gate width
#define DEPTH 10       // reference hardcodes max_depth = 10

__device__ inline v8f wmma4(v2f a, v2f b, v8f c) {
  // D = A(16x4 f32) * B(4x16 f32) + C(16x16 f32), exact f32 MACs.
  return __builtin_amdgcn_wmma_f32_16x16x4_f32(
      /*neg_a=*/false, a, /*neg_b=*/false, b,
      /*c_mod=*/(short)0, c, /*reuse_a=*/false, /*reuse_b=*/false);
}

// K-loop over one (A,B) segment; A,B both row-major [rows, K] (NT GEMM).
// A operand: lane = M row, lane-half selects K pair. B operand: lane = N col.
// RT row sub-tiles x 4 col sub-tiles of 16x16 each.
template <int RT>
__device__ inline void kloop(const float* __restrict__ A, int lda,
                             const float* __restrict__ B, int ldb, int K,
                             const int (&rowA)[RT], const int (&rb)[4],
                             int half, v8f (&acc)[RT][4]) {
  const float* ap[RT];
  const float* bp[4];
#pragma unroll
  for (int r = 0; r < RT; ++r) ap[r] = A + (size_t)rowA[r] * lda + 2 * half;
#pragma unroll
  for (int j = 0; j < 4; ++j) bp[j] = B + (size_t)rb[j] * ldb + 2 * half;
  for (int k = 0; k < K; k += 4) {
    v2f av[RT], bv[4];
#pragma unroll
    for (int r = 0; r < RT; ++r) av[r] = *(const v2f*)(ap[r] + k);
#pragma unroll
    for (int j = 0; j < 4; ++j) bv[j] = *(const v2f*)(bp[j] + k);
#pragma unroll
    for (int r = 0; r < RT; ++r)
#pragma unroll
      for (int j = 0; j < 4; ++j)
        acc[r][j] = wmma4(av[r], bv[j], acc[r][j]);
  }
}

// Generic NT GEMM: C[N,M] = concat_k(A0|A1) * concat_k(B0|B1)^T (+bias)(+tanh)
// A*: [Nrows, K*] row-major, B*: [Mcols, K*] row-major ("weights [out,in]").
// One wave computes a (16*RT)x64 C tile. K0,K1 multiples of 4 (512/1024/2048
// in every call). epilogue: 0 = none, 1 = tanh.
template <int RT>
__global__ __launch_bounds__(128) void gemm_nt_wmma(
    const float* __restrict__ A0, int lda0, const float* __restrict__ B0,
    int ldb0, int K0, const float* __restrict__ A1, int lda1,
    const float* __restrict__ B1, int ldb1, int K1, float* __restrict__ C,
    int ldc, const float* __restrict__ bias, int Nrows, int Mcols,
    int epilogue) {
  const int wave = threadIdx.x >> 5;
  const int lane = threadIdx.x & 31;
  const int half = lane >> 4;
  const int l = lane & 15;

  const int tilesM = (Mcols + 63) >> 6;
  const int tilesN = (Nrows + 16 * RT - 1) / (16 * RT);
  const int tile = blockIdx.x * (blockDim.x >> 5) + wave;
  if (tile >= tilesM * tilesN) return;  // wave-uniform: EXEC stays all-ones
  const int tn = tile / tilesM;
  const int tm = tile % tilesM;

  int rowA[RT];
#pragma unroll
  for (int r = 0; r < RT; ++r) {
    int rr = tn * (16 * RT) + r * 16 + l;
    rowA[r] = (rr < Nrows) ? rr : (Nrows - 1);  // clamp loads; stores masked
  }
  int rb[4];
#pragma unroll
  for (int j = 0; j < 4; ++j) {
    int r = tm * 64 + j * 16 + l;
    rb[j] = (r < Mcols) ? r : (Mcols - 1);
  }

  v8f acc[RT][4];
#pragma unroll
  for (int r = 0; r < RT; ++r)
#pragma unroll
    for (int j = 0; j < 4; ++j) acc[r][j] = (v8f){};

  kloop<RT>(A0, lda0, B0, ldb0, K0, rowA, rb, half, acc);
  if (K1 > 0) kloop<RT>(A1, lda1, B1, ldb1, K1, rowA, rb, half, acc);

  // C/D layout: VGPR i -> row (half*8 + i), lane -> column within tile.
#pragma unroll
  for (int j = 0; j < 4; ++j) {
    int col = tm * 64 + j * 16 + l;
    if (col < Mcols) {
      float bb = bias ? bias[col] : 0.0f;
#pragma unroll
      for (int r = 0; r < RT; ++r) {
#pragma unroll
        for (int i = 0; i < 8; ++i) {
          int row = tn * (16 * RT) + r * 16 + half * 8 + i;
          if (row < Nrows) {
            float v = acc[r][j][i] + bb;
            if (epilogue == 1) v = tanhf(v);
            C[(size_t)row * ldc + col] = v;
          }
        }
      }
    }
  }
}

// annT[h*SD + s] = ann[s*HD + h]  (so ctx = att @ ann becomes an NT GEMM)
__global__ void transpose_ann(const float* __restrict__ ann,
                              float* __restrict__ annT) {
  int idx = blockIdx.x * blockDim.x + threadIdx.x;
  if (idx >= SD * HD) return;
  int s = idx >> 10, h = idx & (HD - 1);
  annT[(size_t)h * SD + s] = ann[idx];
}

__global__ void bias_sum(const float* __restrict__ b_ih,
                         const float* __restrict__ b_hh,
                         float* __restrict__ bsum) {
  int i = blockIdx.x * blockDim.x + threadIdx.x;
  if (i < 2 * GD) bsum[i] = b_ih[i] + b_hh[i];
}

// In-place row softmax over [N, SD]; one block per row.
__global__ __launch_bounds__(256) void softmax_rows(float* __restrict__ att) {
  __shared__ float red[256];
  float* row = att + (size_t)blockIdx.x * SD;
  int t = threadIdx.x;
  float m = -INFINITY;
  for (int c = t; c < SD; c += 256) m = fmaxf(m, row[c]);
  red[t] = m;
  __syncthreads();
  for (int s = 128; s > 0; s >>= 1) {
    if (t < s) red[t] = fmaxf(red[t], red[t + s]);
    __syncthreads();
  }
  m = red[0];
  __syncthreads();
  float sum = 0.0f;
  for (int c = t; c < SD; c += 256) {
    float e = expf(row[c] - m);
    row[c] = e;
    sum += e;
  }
  red[t] = sum;
  __syncthreads();
  for (int s = 128; s > 0; s >>= 1) {
    if (t < s) red[t] += red[t + s];
    __syncthreads();
  }
  float inv = 1.0f / red[0];
  for (int c = t; c < SD; c += 256) row[c] *= inv;
}

// Per row: argmax over VP1 logits (ties -> lowest index, matching jnp.argmax),
// clip to V-1, gather emb[idx] into xh[:, 0:HD]. One block per row.
__global__ __launch_bounds__(256) void argmax_gather(
    const float* __restrict__ logits, const float* __restrict__ emb,
    float* __restrict__ xh) {
  __shared__ float vmax[256];
  __shared__ int vidx[256];
  int n = blockIdx.x, t = threadIdx.x;
  const float* row = logits + (size_t)n * VP1;
  float bm = -INFINITY;
  int bi = 0;
  for (int c = t; c < VP1; c += 256) {
    float v = row[c];
    if (v > bm) { bm = v; bi = c; }
  }
  vmax[t] = bm;
  vidx[t] = bi;
  __syncthreads();
  for (int s = 128; s > 0; s >>= 1) {
    if (t < s) {
      if (vmax[t + s] > vmax[t] ||
          (vmax[t + s] == vmax[t] && vidx[t + s] < vidx[t])) {
        vmax[t] = vmax[t + s];
        vidx[t] = vidx[t + s];
      }
    }
    __syncthreads();
  }
  int idx = vidx[0];
  if (idx > SD - 1) idx = SD - 1;  // clip(pred, 0, V-1), V == 512
  const float* e = emb + (size_t)idx * HD;
  float* dst = xh + (size_t)n * (2 * HD);
  for (int c = t; c < HD; c += 256) dst[c] = e[c];
}

// LSTM pointwise for one child k: gates[N,GD] (order i,f,g,o), Cs[N,HD] ->
// Hn/Cn rows [k*N .. k*N+N-1].
__global__ void lstm_pointwise(const float* __restrict__ gates,
                               const float* __restrict__ Cs,
                               float* __restrict__ Hn, float* __restrict__ Cn,
                               int N, int k) {
  int idx = blockIdx.x * blockDim.x + threadIdx.x;
  if (idx >= N * HD) return;
  int c = idx & (HD - 1);
  int n = idx >> 10;
  const float* g = gates + (size_t)n * GD;
  float i_ = g[c], f_ = g[HD + c], gg = g[2 * HD + c], o_ = g[3 * HD + c];
  float si = 1.0f / (1.0f + expf(-i_));
  float sf = 1.0f / (1.0f + expf(-f_));
  float so = 1.0f / (1.0f + expf(-o_));
  float nc = sf * Cs[(size_t)n * HD + c] + si * tanhf(gg);
  size_t orow = (size_t)(k * N + n) * HD + c;
  Cn[orow] = nc;
  Hn[orow] = so * tanhf(nc);
}

// ---------------------------------------------------------------------------

static inline void launch_gemm(hipStream_t stream, const float* A0, int lda0,
                               const float* B0, int ldb0, int K0,
                               const float* A1, int lda1, const float* B1,
                               int ldb1, int K1, float* C, int ldc,
                               const float* bias, int Nrows, int Mcols,
                               int ep) {
  int tilesM = (Mcols + 63) / 64;
  if (Nrows >= 64) {
    // 64x64 C tile per wave: 16 WMMA per 8 loads per k-step.
    int tiles = ((Nrows + 63) / 64) * tilesM;
    int blocks = (tiles + 3) / 4;  // 4 waves / block
    gemm_nt_wmma<4><<<blocks, 128, 0, stream>>>(A0, lda0, B0, ldb0, K0, A1,
                                                lda1, B1, ldb1, K1, C, ldc,
                                                bias, Nrows, Mcols, ep);
  } else {
    int tiles = ((Nrows + 15) / 16) * tilesM;
    int blocks = (tiles + 3) / 4;
    gemm_nt_wmma<1><<<blocks, 128, 0, stream>>>(A0, lda0, B0, ldb0, K0, A1,
                                                lda1, B1, ldb1, K1, C, ldc,
                                                bias, Nrows, Mcols, ep);
  }
}

extern "C" void kernel_launch(void* const* d_in, const int* in_sizes, int n_in,
                              void* d_out, int out_size, void* d_ws,
                              size_t ws_size, hipStream_t stream) {
  (void)in_sizes; (void)n_in; (void)out_size; (void)ws_size;
  const float* rootH  = (const float*)d_in[0];
  const float* rootC  = (const float*)d_in[1];
  const float* ann    = (const float*)d_in[2];
  const float* W_ah   = (const float*)d_in[3];
  const float* b_ah   = (const float*)d_in[4];
  const float* W_pre  = (const float*)d_in[5];   // [HD, 2*HD]
  const float* b_pre  = (const float*)d_in[6];
  const float* W_pred = (const float*)d_in[7];   // [VP1, HD]
  const float* emb    = (const float*)d_in[8];   // [SD, HD]
  const float* W_ih   = (const float*)d_in[9];   // [2, GD, 2*HD]
  const float* W_hh   = (const float*)d_in[10];  // [2, GD, HD]
  const float* b_ih   = (const float*)d_in[11];
  const float* b_hh   = (const float*)d_in[12];
  float* out = (float*)d_out;

  // Workspace layout (floats); total ~13.1M floats (~53 MB).
  float* ws    = (float*)d_ws;
  float* hv    = ws;                    // [SD, HD]
  float* annT  = hv + SD * HD;          // [HD, SD]
  float* HsA   = annT + HD * SD;        // [1024, HD]
  float* CsA   = HsA + 1024 * HD;
  float* HsB   = CsA + 1024 * HD;
  float* CsB   = HsB + 1024 * HD;
  float* att   = CsB + 1024 * HD;       // [1024, SD]
  float* ctx   = att + 1024 * SD;       // [1024, HD]
  float* xh    = ctx + 1024 * HD;       // [1024, 2*HD]: [emb | et]
  float* gates = xh + 1024 * 2 * HD;    // [1024, GD] (per child k)
  float* bsum  = gates + 1024 * GD;     // [2, GD]

  // ---- per-launch precompute ----
  hipMemcpyAsync(HsA, rootH, HD * sizeof(float), hipMemcpyDeviceToDevice,
                 stream);
  hipMemcpyAsync(CsA, rootC, HD * sizeof(float), hipMemcpyDeviceToDevice,
                 stream);
  transpose_ann<<<(SD * HD + 255) / 256, 256, 0, stream>>>(ann, annT);
  bias_sum<<<(2 * GD + 255) / 256, 256, 0, stream>>>(b_ih, b_hh, bsum);
  // hv = ann @ W_ah^T + b_ah   [SD, HD]
  launch_gemm(stream, ann, HD, W_ah, HD, HD, nullptr, 0, nullptr, 0, 0, hv,
              HD, b_ah, SD, HD, 0);

  float* Hs = HsA; float* Cs = CsA; float* Hn = HsB; float* Cn = CsB;
  for (int d = 0; d <= DEPTH; ++d) {
    int N = 1 << d;
    int off = N - 1;  // output row offset for this level
    // att = Hs @ hv^T                            [N, SD]
    launch_gemm(stream, Hs, HD, hv, HD, HD, nullptr, 0, nullptr, 0, 0, att,
                SD, nullptr, N, SD, 0);
    softmax_rows<<<N, 256, 0, stream>>>(att);
    // ctx = att @ ann = att @ annT^T             [N, HD]
    launch_gemm(stream, att, SD, annT, SD, SD, nullptr, 0, nullptr, 0, 0, ctx,
                HD, nullptr, N, HD, 0);
    // et = tanh([Hs|ctx] @ W_pre^T + b_pre) -> xh[:, HD:2HD]
    launch_gemm(stream, Hs, HD, W_pre, 2 * HD, HD, ctx, HD, W_pre + HD,
                2 * HD, HD, xh + HD, 2 * HD, b_pre, N, HD, 1);
    // logits = et @ W_pred^T -> out rows [off .. off+N)
    launch_gemm(stream, xh + HD, 2 * HD, W_pred, HD, HD, nullptr, 0, nullptr,
                0, 0, out + (size_t)off * VP1, VP1, nullptr, N, VP1, 0);
    if (d == DEPTH) break;
    // pred = argmax(logits); xh[:, 0:HD] = emb[clip(pred)]
    argmax_gather<<<N, 256, 0, stream>>>(out + (size_t)off * VP1, emb, xh);
    for (int k = 0; k < 2; ++k) {
      // gates = [x|Hs] @ [W_ih[k]|W_hh[k]]^T + (b_ih[k]+b_hh[k])   [N, GD]
      launch_gemm(stream, xh, 2 * HD, W_ih + (size_t)k * GD * 2 * HD, 2 * HD,
                  2 * HD, Hs, HD, W_hh + (size_t)k * GD * HD, HD, HD, gates,
                  GD, bsum + k * GD, N, GD, 0);
      lstm_pointwise<<<(N * HD + 255) / 256, 256, 0, stream>>>(gates, Cs, Hn,
                                                               Cn, N, k);
    }
    float* t;
    t = Hs; Hs = Hn; Hn = t;
    t = Cs; Cs = Cn; Cn = t;
  }
}